// Multislice_47072841564280
// MI455X (gfx1250) — compile-verified
//
#include <hip/hip_runtime.h>
#include <math.h>

// ---------------------------------------------------------------------------
// Multislice gradient step on MI455X (gfx1250, wave32).
//
// prop(u,kern) = ifft2(fft2(u)*kern) is implemented as 5 dense passes:
//   rowFFT_fwd(+fused premul) -> transpose -> [colFFT_fwd * kern * colFFT_inv
//   fused in LDS] -> transpose -> rowFFT_inv(+fused postmul/div).
// The 1024-pt FFT = 16 x FFT64 (Stockham radix-2 in LDS, stride-65 padded)
// + twiddle + radix-16 DFT executed as V_WMMA_F32_16X16X4_F32 chains with
// the DFT-matrix operands precomputed into registers (pinned with empty
// inline asm so the scheduler cannot rematerialize sin/cos into the chain).
// ---------------------------------------------------------------------------

#define FN   1024
#define FNZ  16

#define PIF      3.14159265358979323846f
#define TWO_PIF  6.28318530717958647692f
#define PS_EFF   0.325f                 // 6.5 / 20
#define WLEN     0.5f
#define DZS      3.25f                  // 10 * 6.5 / 20
#define KAPPA    (TWO_PIF * DZS / WLEN) // ~40.84
#define DFX      (1.0f / (PS_EFF * (float)FN))
#define NA_CUT   1.69f                  // (0.65/0.5)^2
#define INV_WL2  4.0f                   // (1/0.5)^2
#define FOCUS_DZ (-DZS * (float)(FNZ - 1) * 0.5f)

typedef float v2f __attribute__((ext_vector_type(2)));
typedef float v8f __attribute__((ext_vector_type(8)));

#if __has_builtin(__builtin_amdgcn_wmma_f32_16x16x4_f32)
#define USE_WMMA_F32 1
#else
#define USE_WMMA_F32 0
#endif

// Force a float to live in a VGPR; blocks rematerialization/sinking of the
// computation that produced it (device pass only).
#if defined(__HIP_DEVICE_COMPILE__)
#define PIN_VGPR(x) asm("" : "+v"(x))
#else
#define PIN_VGPR(x) ((void)(x))
#endif

// ----------------------------- complex helpers -----------------------------
__device__ __forceinline__ float2 cmul2(float2 a, float2 b) {
  return make_float2(a.x * b.x - a.y * b.y, a.x * b.y + a.y * b.x);
}
__device__ __forceinline__ float2 cmulc2(float2 a, float2 b) { // a * conj(b)
  return make_float2(a.x * b.x + a.y * b.y, a.y * b.x - a.x * b.y);
}
__device__ __forceinline__ float2 cdiv2(float2 a, float2 b) { // a / b
  float inv = 1.0f / (b.x * b.x + b.y * b.y);
  return make_float2((a.x * b.x + a.y * b.y) * inv, (a.y * b.x - a.x * b.y) * inv);
}

// --------------------- FFT building blocks (in LDS) ------------------------
// 16 independent 64-pt Stockham radix-2 FFTs. Rows padded to stride 65
// (64 LDS banks -> conflict free). Result ends in (xr,xi) = first buffer.
__device__ __forceinline__ void fft64x16(float* xr, float* xi, float* yr,
                                         float* yi, float sgn, int tid) {
  int ncur = 64, s = 1;
#pragma unroll
  for (int st = 0; st < 6; ++st) {
    const int m = ncur >> 1;
#pragma unroll
    for (int half = 0; half < 2; ++half) {
      int w   = tid + (half << 8);        // 0..511
      int row = w >> 5;                   // 0..15
      int idx = w & 31;                   // m*s == 32 always
      int p   = idx / s;
      int q   = idx - p * s;
      float th = sgn * TWO_PIF * (float)p / (float)ncur;
      float ws, wc;
      __sincosf(th, &ws, &wc);
      int b0 = row * 65;
      float ar = xr[b0 + q + s * p],       ai = xi[b0 + q + s * p];
      float br = xr[b0 + q + s * (p + m)], bi = xi[b0 + q + s * (p + m)];
      yr[b0 + q + s * (2 * p)]     = ar + br;
      yi[b0 + q + s * (2 * p)]     = ai + bi;
      float dr = ar - br, di = ai - bi;
      yr[b0 + q + s * (2 * p + 1)] = dr * wc - di * ws;
      yi[b0 + q + s * (2 * p + 1)] = dr * ws + di * wc;
    }
    __syncthreads();
    float* t;
    t = xr; xr = yr; yr = t;
    t = xi; xi = yi; yi = t;
    ncur = m; s <<= 1;
  }
}

// Y[n1][k2] *= exp(sgn*i*2*pi*n1*k2/1024)
__device__ __forceinline__ void twiddle1024(float* re, float* im, float sgn,
                                            int tid) {
  for (int e = tid; e < 1024; e += 256) {
    int n1 = e >> 6, k2 = e & 63;
    float th = sgn * TWO_PIF * (float)(n1 * k2) * (1.0f / 1024.0f);
    float s, c;
    __sincosf(th, &s, &c);
    int idx = n1 * 65 + k2;
    float r = re[idx], i = im[idx];
    re[idx] = r * c - i * s;
    im[idx] = r * s + i * c;
  }
  __syncthreads();
}

// Per-lane DFT-matrix constants for the radix-16 WMMA stage, computed once
// per kernel.  fc = cos entries (A layout: lane = M + 16*(K/2), vgpr = K&1),
// fsg = sin entries pre-multiplied by (isImag ? +1 : -1); multiplying fsg by
// the FFT direction sgn yields the sine-term A operand.
__device__ __forceinline__ void radix16_prep(int tid, float* fc, float* fsg) {
#if USE_WMMA_F32
  const int lane   = tid & 31;
  const int wave   = tid >> 5;
  const int isImag = wave >> 2;
  const int m      = lane & 15;
  const int kb     = (lane >> 4) << 1;
  const float sflip = isImag ? 1.0f : -1.0f;
#pragma unroll
  for (int kk = 0; kk < 4; ++kk) {
#pragma unroll
    for (int j = 0; j < 2; ++j) {
      int k = 4 * kk + kb + j;
      float s, c;
      __sincosf((float)((m * k) & 15) * (PIF / 8.0f), &s, &c);
      fc[2 * kk + j]  = c;
      fsg[2 * kk + j] = sflip * s;
    }
  }
#else
  (void)tid; (void)fc; (void)fsg;
#endif
}

// Radix-16 stage on tensor cores: C[k1,k2] = sum_n1 F16[k1,n1] * Y[n1,k2],
// F16 = exp(sgn*i*2*pi*k1*n1/16).  Complex matmul as two real WMMA chains:
//   waves 0-3 (Cr): cos-term from reA, sin-term (-sgn*sin) from imA
//   waves 4-7 (Ci): cos-term from imA, sin-term (+sgn*sin) from reA
// B-source pointers selected once per wave; inner loop is ds_load + wmma.
// Input: (reA,imA) stride-65 matrix.  Output: (reB,imB) flat natural order
// k = k2 + 64*k1.  All 8 waves fully active (EXEC all ones).
__device__ __forceinline__ void radix16_stage(const float* reA,
                                              const float* imA, float* reB,
                                              float* imB, float sgn, int tid,
                                              const float* fc,
                                              const float* fss) {
#if USE_WMMA_F32
  (void)sgn;
  const int lane    = tid & 31;
  const int wave    = tid >> 5;
  const int t       = wave & 3;           // k2 tile (16 wide)
  const int isImag  = wave >> 2;          // 0 = real part, 1 = imag part
  const int kb      = (lane >> 4) << 1;   // B K sub-offset per ISA layout
  const int colBase = 16 * t + (lane & 15);
  const float* BsrcCos = isImag ? imA : reA;
  const float* BsrcSin = isImag ? reA : imA;
  v8f acc = {0.f, 0.f, 0.f, 0.f, 0.f, 0.f, 0.f, 0.f};
#pragma unroll
  for (int kk = 0; kk < 4; ++kk) {        // cosine term
    int k0 = 4 * kk + kb;
    v2f a, b;
    a.x = fc[2 * kk];
    a.y = fc[2 * kk + 1];
    b.x = BsrcCos[(k0 + 0) * 65 + colBase];
    b.y = BsrcCos[(k0 + 1) * 65 + colBase];
    acc = __builtin_amdgcn_wmma_f32_16x16x4_f32(false, a, false, b, (short)0,
                                                acc, false, false);
  }
#pragma unroll
  for (int kk = 0; kk < 4; ++kk) {        // sine term
    int k0 = 4 * kk + kb;
    v2f a, b;
    a.x = fss[2 * kk];
    a.y = fss[2 * kk + 1];
    b.x = BsrcSin[(k0 + 0) * 65 + colBase];
    b.y = BsrcSin[(k0 + 1) * 65 + colBase];
    acc = __builtin_amdgcn_wmma_f32_16x16x4_f32(false, a, false, b, (short)0,
                                                acc, false, false);
  }
  float* dst = isImag ? imB : reB;
  int mhi = (lane >> 4) << 3;             // C layout: VGPR r -> M = r (+8)
#pragma unroll
  for (int r = 0; r < 8; ++r) {
    int k1 = r + mhi;
    dst[colBase + (k1 << 6)] = acc[r];
  }
  __syncthreads();
#else
  (void)fc; (void)fss;
  for (int e = tid; e < 1024; e += 256) {
    int k1 = e >> 6, k2 = e & 63;
    float sr = 0.f, si = 0.f;
#pragma unroll
    for (int n1 = 0; n1 < 16; ++n1) {
      float s, c;
      __sincosf((float)((k1 * n1) & 15) * (PIF / 8.0f), &s, &c);
      s *= sgn;
      float yr = reA[n1 * 65 + k2], yi = imA[n1 * 65 + k2];
      sr += c * yr - s * yi;
      si += c * yi + s * yr;
    }
    reB[e] = sr;
    imB[e] = si;
  }
  __syncthreads();
#endif
}

// ------------------------------ FFT kernels --------------------------------
// One 1024-pt FFT per 256-thread block (8 waves).  preMode: 0 none,
// 1 v*=pre, 2 v*=conj(pre).  postMode: 0 none, 1 v*=post, 2 v/=post.
__global__ __launch_bounds__(256) void fft_row(
    const float2* __restrict__ in, float2* __restrict__ out,
    const float2* __restrict__ pre, int preMode,
    const float2* __restrict__ post, int postMode, float sgn, float scale) {
  __shared__ float reA[1040], imA[1040], reB[1040], imB[1040];
  const int tid = threadIdx.x;
  const size_t base = (size_t)blockIdx.x * FN;

  if (pre)  __builtin_prefetch(&pre[base + tid * 4], 0, 1);
  if (post) __builtin_prefetch(&post[base + tid * 4], 0, 1);

  float fc[8], fsg[8], fss[8];
  radix16_prep(tid, fc, fsg);
#pragma unroll
  for (int j = 0; j < 8; ++j) {
    fss[j] = sgn * fsg[j];
    PIN_VGPR(fc[j]);
    PIN_VGPR(fss[j]);
  }

  for (int e = tid; e < 1024; e += 256) {
    float2 v = in[base + e];
    if (preMode == 1)      v = cmul2(v, pre[base + e]);
    else if (preMode == 2) v = cmulc2(v, pre[base + e]);
    int n1 = e & 15, n2 = e >> 4;         // x[n1 + 16*n2] -> Y0[n1][n2]
    reA[n1 * 65 + n2] = v.x;
    imA[n1 * 65 + n2] = v.y;
  }
  __syncthreads();
  fft64x16(reA, imA, reB, imB, sgn, tid);
  twiddle1024(reA, imA, sgn, tid);
  radix16_stage(reA, imA, reB, imB, sgn, tid, fc, fss);
  for (int e = tid; e < 1024; e += 256) {
    float2 v = make_float2(reB[e] * scale, imB[e] * scale);
    if (postMode == 1)      v = cmul2(v, post[base + e]);
    else if (postMode == 2) v = cdiv2(v, post[base + e]);
    out[base + e] = v;
  }
}

// Fused column pass (rows are pre-transposed): fwd FFT, multiply by the
// transpose-invariant propagation kernel, inverse FFT with 1/1024 scaling.
__global__ __launch_bounds__(256) void fft_row_fused(
    const float2* __restrict__ in, float2* __restrict__ out,
    const float2* __restrict__ kern, int conjK) {
  __shared__ float reA[1040], imA[1040], reB[1040], imB[1040];
  const int tid = threadIdx.x;
  const size_t base = (size_t)blockIdx.x * FN;

  __builtin_prefetch(&kern[base + tid * 4], 0, 1);

  float fc[8], fsg[8], fssF[8], fssI[8];
  radix16_prep(tid, fc, fsg);
#pragma unroll
  for (int j = 0; j < 8; ++j) {
    fssF[j] = -fsg[j];
    fssI[j] = fsg[j];
    PIN_VGPR(fc[j]);
    PIN_VGPR(fssF[j]);
    PIN_VGPR(fssI[j]);
  }

  for (int e = tid; e < 1024; e += 256) {
    float2 v = in[base + e];
    int n1 = e & 15, n2 = e >> 4;
    reA[n1 * 65 + n2] = v.x;
    imA[n1 * 65 + n2] = v.y;
  }
  __syncthreads();
  fft64x16(reA, imA, reB, imB, -1.0f, tid);
  twiddle1024(reA, imA, -1.0f, tid);
  radix16_stage(reA, imA, reB, imB, -1.0f, tid, fc, fssF); // spectrum in B
  // kernel multiply fused with the re-permute that feeds the inverse FFT
  for (int e = tid; e < 1024; e += 256) {
    float2 v = make_float2(reB[e], imB[e]);
    float2 k = kern[base + e];
    if (conjK) k.y = -k.y;
    v = cmul2(v, k);
    int n1 = e & 15, n2 = e >> 4;
    reA[n1 * 65 + n2] = v.x;
    imA[n1 * 65 + n2] = v.y;
  }
  __syncthreads();
  fft64x16(reA, imA, reB, imB, 1.0f, tid);
  twiddle1024(reA, imA, 1.0f, tid);
  radix16_stage(reA, imA, reB, imB, 1.0f, tid, fc, fssI);
  const float scale = 1.0f / 1024.0f;
  for (int e = tid; e < 1024; e += 256) {
    out[base + e] = make_float2(reB[e] * scale, imB[e] * scale);
  }
}

// 32x32 tiled transpose (dense both ways).
__global__ __launch_bounds__(256) void transpose_k(
    const float2* __restrict__ in, float2* __restrict__ out) {
  __shared__ float2 tile[32][33];
  int bx = blockIdx.x * 32, by = blockIdx.y * 32;
  int tx = threadIdx.x, ty = threadIdx.y;
#pragma unroll
  for (int j = 0; j < 32; j += 8)
    tile[ty + j][tx] = in[(size_t)(by + ty + j) * FN + bx + tx];
  __syncthreads();
#pragma unroll
  for (int j = 0; j < 32; j += 8)
    out[(size_t)(bx + ty + j) * FN + by + tx] = tile[tx][ty + j];
}

// --------------------------- elementwise kernels ---------------------------
// T[z][pix] = exp(i*kappa*(Or + i*Oi)) = exp(-kappa*Oi)*(cos+isin)(kappa*Or)
__global__ void build_T(const float* __restrict__ x, float2* __restrict__ T) {
  int p = blockIdx.x * blockDim.x + threadIdx.x;  // z*NN + pix
  int pix = p & (FN * FN - 1);
  int z   = p >> 20;
  float Or = x[(size_t)pix * 2 * FNZ + z];
  float Oi = x[(size_t)pix * 2 * FNZ + FNZ + z];
  float amp = __expf(-KAPPA * Oi);
  float s, c;
  __sincosf(KAPPA * Or, &s, &c);
  T[p] = make_float2(amp * c, amp * s);
}

// Band-limited angular-spectrum kernels + pupil, in fft order.
__global__ void build_kernels(float2* __restrict__ kP, float2* __restrict__ kF,
                              float2* __restrict__ pup) {
  int p = blockIdx.x * blockDim.x + threadIdx.x;
  int kx = p & (FN - 1), ky = p >> 10;
  float fx = (float)((kx < 512) ? kx : kx - 1024) * DFX;
  float fy = (float)((ky < 512) ? ky : ky - 1024) * DFX;
  float urr = fx * fx + fy * fy;
  float pc = (urr < NA_CUT) ? 1.0f : 0.0f;
  float kz = sqrtf(fmaxf(pc * (INV_WL2 - urr), 0.0f));
  float s, c;
  __sincosf(TWO_PIF * DZS * kz, &s, &c);
  kP[p] = make_float2(pc * c, pc * s);
  __sincosf(TWO_PIF * FOCUS_DZ * kz, &s, &c);
  kF[p] = make_float2(pc * c, pc * s);
  pup[p] = make_float2(pc, 0.0f);
}

__global__ void plane_wave(const float* __restrict__ na,
                           float2* __restrict__ u) {
  int p = blockIdx.x * blockDim.x + threadIdx.x;
  int xc = p & (FN - 1), yc = p >> 10;
  float tx = roundf((na[0] / WLEN) / DFX) * DFX;
  float ty = roundf((na[1] / WLEN) / DFX) * DFX;
  float X = (float)(xc - 512) * PS_EFF;
  float Y = (float)(yc - 512) * PS_EFF;
  float ph = TWO_PIF * (tx * X + ty * Y);
  float s, c;
  __sincosf(ph, &s, &c);
  u[p] = make_float2(c, s);
}

__global__ void cmul_k(float2* __restrict__ a, const float2* __restrict__ b) {
  int p = blockIdx.x * blockDim.x + threadIdx.x;
  a[p] = cmul2(a[p], b[p]);
}

// u -= sqrt(meas) * u / |u|
__global__ void residual_k(float2* __restrict__ u,
                           const float* __restrict__ meas) {
  int p = blockIdx.x * blockDim.x + threadIdx.x;
  float2 v = u[p];
  float mag = sqrtf(v.x * v.x + v.y * v.y);
  float f = 1.0f - sqrtf(meas[p]) / mag;
  u[p] = make_float2(v.x * f, v.y * f);
}

// G = u_bp * conj(u) * conj(T) * conj(i*kappa)   (conj(K) = -i*kappa)
__global__ void grad_k(const float2* __restrict__ ubp,
                       const float2* __restrict__ u,
                       const float2* __restrict__ Tz, float2* __restrict__ G) {
  int p = blockIdx.x * blockDim.x + threadIdx.x;
  float2 w = cmulc2(ubp[p], u[p]);
  w = cmulc2(w, Tz[p]);
  G[p] = make_float2(KAPPA * w.y, -KAPPA * w.x);
}

// out[N,N,2,NZ] = x - alpha * G  (dense 128B read+write per pixel)
__global__ void interleave_k(const float* __restrict__ x,
                             const float* __restrict__ alpha,
                             const float2* __restrict__ G,
                             float* __restrict__ out) {
  int pix = blockIdx.x * blockDim.x + threadIdx.x;
  float a = alpha[0];
  size_t b = (size_t)pix * 2 * FNZ;
#pragma unroll
  for (int z = 0; z < FNZ; ++z) {
    float2 g = G[(size_t)z * FN * FN + pix];
    out[b + z]       = x[b + z] - a * g.x;
    out[b + FNZ + z] = x[b + FNZ + z] - a * g.y;
  }
}

// ------------------------------ orchestration ------------------------------
extern "C" void kernel_launch(void* const* d_in, const int* in_sizes, int n_in,
                              void* d_out, int out_size, void* d_ws,
                              size_t ws_size, hipStream_t stream) {
  (void)in_sizes; (void)n_in; (void)out_size; (void)ws_size;
  const float* x     = (const float*)d_in[0];
  const float* meas  = (const float*)d_in[1];
  const float* na    = (const float*)d_in[2];
  const float* alpha = (const float*)d_in[3];
  float* out = (float*)d_out;

  const size_t NN = (size_t)FN * FN;
  char* w = (char*)d_ws;
  auto take = [&](size_t bytes) { char* p = w; w += bytes; return p; };
  float2* T   = (float2*)take(NN * FNZ * sizeof(float2));
  float2* G   = (float2*)take(NN * FNZ * sizeof(float2));
  float2* kP  = (float2*)take(NN * sizeof(float2));
  float2* kF  = (float2*)take(NN * sizeof(float2));
  float2* pup = (float2*)take(NN * sizeof(float2));
  float2* u   = (float2*)take(NN * sizeof(float2));
  float2* ubp = (float2*)take(NN * sizeof(float2));
  float2* tA  = (float2*)take(NN * sizeof(float2));
  float2* tB  = (float2*)take(NN * sizeof(float2));

  const dim3 eBlk(256), eGrd(NN / 256), zGrd(NN * FNZ / 256);
  const dim3 trBlk(32, 8), trGrd(32, 32);

  auto do_prop = [&](const float2* in, float2* outp, const float2* pre,
                     int preMode, const float2* kern, int conjK,
                     const float2* post, int postMode) {
    fft_row<<<FN, 256, 0, stream>>>(in, tA, pre, preMode, nullptr, 0, -1.0f,
                                    1.0f);
    transpose_k<<<trGrd, trBlk, 0, stream>>>(tA, tB);
    fft_row_fused<<<FN, 256, 0, stream>>>(tB, tA, kern, conjK);
    transpose_k<<<trGrd, trBlk, 0, stream>>>(tA, tB);
    fft_row<<<FN, 256, 0, stream>>>(tB, outp, nullptr, 0, post, postMode, 1.0f,
                                    1.0f / 1024.0f);
  };

  build_T<<<zGrd, eBlk, 0, stream>>>(x, T);
  build_kernels<<<eGrd, eBlk, 0, stream>>>(kP, kF, pup);
  plane_wave<<<eGrd, eBlk, 0, stream>>>(na, u);

  // forward multislice: u = prop(T_z * u) for z = 0..14, then u *= T_15
  for (int z = 0; z < FNZ - 1; ++z)
    do_prop(u, u, T + (size_t)z * NN, 1, kP, 0, nullptr, 0);
  cmul_k<<<eGrd, eBlk, 0, stream>>>(u, T + (size_t)(FNZ - 1) * NN);

  // refocus + pupil (keep u intact for the backward recursion)
  do_prop(u, ubp, nullptr, 0, kF, 0, nullptr, 0);
  do_prop(ubp, ubp, nullptr, 0, pup, 0, nullptr, 0);
  residual_k<<<eGrd, eBlk, 0, stream>>>(ubp, meas);
  do_prop(ubp, ubp, nullptr, 0, pup, 1, nullptr, 0);
  do_prop(ubp, ubp, nullptr, 0, kF, 1, nullptr, 0);

  // backward slice recursion
  for (int zz = FNZ - 1; zz >= 0; --zz) {
    grad_k<<<eGrd, eBlk, 0, stream>>>(ubp, u, T + (size_t)zz * NN,
                                      G + (size_t)zz * NN);
    if (zz > 0) {
      do_prop(ubp, ubp, T + (size_t)zz * NN, 2, kP, 1, nullptr, 0);
      do_prop(u, u, nullptr, 0, kP, 1, T + (size_t)(zz - 1) * NN, 2);
    }
  }

  interleave_k<<<eGrd, eBlk, 0, stream>>>(x, alpha, G, out);
}